// HiPPOScaleFast_65008624992666
// MI455X (gfx1250) — compile-verified
//
#include <hip/hip_runtime.h>
#include <hip/hip_bf16.h>

// HiPPO-LegS scan  x_t = A_t x_{t-1} + B_t*u_t  as a 3-phase chunked parallel
// scan on gfx1250.  State X is kept as a (N=64 rows) x (32 cols) register tile
// per wave in WMMA B-operand layout; each step is K=64 of f32 WMMA 16x16x4.
// A_t is staged into LDS per WG with double-buffered async loads
// (global_load_async_to_lds_b128 / ASYNCcnt), pitch-padded to kill bank
// conflicts, so the 8 waves of a WG read each A_t from LDS instead of 8x
// redundant global traffic.
//
//   phaseP : P_c = A_{c,last} ... A_{c,first}          (64 WGs x 2 waves)
//   phaseA : y_last(c) = local scan, zero init         (64 WGs x 8 waves)
//   phaseB : s_{c+1} = P_c s_c + y_last(c)   (serial)  (1 WG  x 8 waves)
//   phaseC : full scan per chunk from s_c -> d_out     (64 WGs x 8 waves)

#define LSEQ   4096
#define NB     256
#define NST    64
#define CHUNKS 64
#define TCH    (LSEQ / CHUNKS)   // 64
#define APITCH 68                // LDS row pitch in floats (272B, 16B aligned)
#define ABUFB  (NST * APITCH * 4)  // bytes per LDS A buffer (17408)

typedef __attribute__((ext_vector_type(2))) float v2f;
typedef __attribute__((ext_vector_type(4))) float v4f;
typedef __attribute__((ext_vector_type(8))) float v8f;

// SWAPX16: exchange data between lanes 0-15 and 16-31 (wave32 halves).
__device__ __forceinline__ float swap16(float x) {
  return __builtin_bit_cast(
      float, __builtin_amdgcn_ds_swizzle(__builtin_bit_cast(int, x), 0x401f));
}

// Stage one 64x64 f32 A_t (16KB) into LDS at byte offset `ldsbase`, padded to
// APITCH floats per row.  1024 16-byte chunks split across NT threads; each
// lane issues async b128 copies tracked on ASYNCcnt.
template <int NT>
__device__ __forceinline__ void stage_A_async(const float* __restrict__ gbase,
                                              unsigned ldsbase, int tid) {
#pragma unroll
  for (int k = 0; k < 1024 / NT; ++k) {
    int idx = tid + k * NT;  // 16B chunk index; 16 chunks per 64-float row
    unsigned lds_addr =
        ldsbase + (unsigned)((idx >> 4) * (APITCH * 4) + (idx & 15) * 16);
    unsigned voff = (unsigned)(idx * 16);
    asm volatile("global_load_async_to_lds_b128 %0, %1, %2"
                 :
                 : "v"(lds_addr), "v"(voff), "s"(gbase)
                 : "memory");
  }
}

__device__ __forceinline__ void wait_async_all() {
  asm volatile("s_wait_asynccnt 0" ::: "memory");
}

// One recurrence step:  D[ct][rt] += A_t(16x4 frags) * Xb(4x16 frags), K = 64.
// A-frag layout (ISA 7.12.2): lanes0-15 M=lane, VGPR0/1 = K,K+1;
//                             lanes16-31 M=lane-16, VGPR0/1 = K+2,K+3.
// A fragments read from the padded LDS tile (ds_load_b64, conflict-free).
__device__ __forceinline__ void scan_step_lds(const float* Ash, v2f Xb[2][16],
                                              v8f D[2][4], int l15, int kof) {
#pragma unroll
  for (int kc = 0; kc < 16; ++kc) {
    v2f af[4];
#pragma unroll
    for (int rt = 0; rt < 4; ++rt)
      af[rt] = *(const v2f*)(Ash + (rt * 16 + l15) * APITCH + kc * 4 + kof);
#pragma unroll
    for (int rt = 0; rt < 4; ++rt)
#pragma unroll
      for (int ct = 0; ct < 2; ++ct)
        D[ct][rt] = __builtin_amdgcn_wmma_f32_16x16x4_f32(
            false, af[rt], false, Xb[ct][kc], (short)0, D[ct][rt], false,
            false);
  }
}

// Same, reading A fragments straight from global (phase B on ws_P).
__device__ __forceinline__ void scan_step(const float* __restrict__ Ap,
                                          v2f Xb[2][16], v8f D[2][4], int l15,
                                          int kof) {
#pragma unroll
  for (int kc = 0; kc < 16; ++kc) {
    v2f af[4];
#pragma unroll
    for (int rt = 0; rt < 4; ++rt)
      af[rt] = *(const v2f*)(Ap + (rt * 16 + l15) * NST + kc * 4 + kof);
#pragma unroll
    for (int rt = 0; rt < 4; ++rt)
#pragma unroll
      for (int ct = 0; ct < 2; ++ct)
        D[ct][rt] = __builtin_amdgcn_wmma_f32_16x16x4_f32(
            false, af[rt], false, Xb[ct][kc], (short)0, D[ct][rt], false,
            false);
  }
}

// Re-layout new state from C/D tiles (VGPR r: rows M=r | M=r+8 in lane halves)
// into B-operand K-fragments (VGPR v: rows K | K+2 in lane halves) using
// SWAPX16 + per-half selects.  Pure register traffic, no barriers.
__device__ __forceinline__ void d_to_x(v8f D[2][4], v2f Xb[2][16], bool low) {
#pragma unroll
  for (int ct = 0; ct < 2; ++ct)
#pragma unroll
    for (int rt = 0; rt < 4; ++rt) {
      v8f d = D[ct][rt];
      v2f f0, f1, f2, f3;
      f0.x = low ? d[0] : swap16(d[2]);
      f0.y = low ? d[1] : swap16(d[3]);
      f1.x = low ? d[4] : swap16(d[6]);
      f1.y = low ? d[5] : swap16(d[7]);
      f2.x = low ? swap16(d[0]) : d[2];
      f2.y = low ? swap16(d[1]) : d[3];
      f3.x = low ? swap16(d[4]) : d[6];
      f3.y = low ? swap16(d[5]) : d[7];
      Xb[ct][4 * rt + 0] = f0;
      Xb[ct][4 * rt + 1] = f1;
      Xb[ct][4 * rt + 2] = f2;
      Xb[ct][4 * rt + 3] = f3;
    }
}

// Rank-1 injection: D = outer(B_t, u_t) directly in C/D layout.
__device__ __forceinline__ void init_u(const float* __restrict__ inp,
                                       const float* __restrict__ Bv, int t,
                                       int colbase, int l15, bool low,
                                       v8f D[2][4]) {
#pragma unroll
  for (int ct = 0; ct < 2; ++ct) {
    float u = inp[t * NB + colbase + ct * 16 + l15];
#pragma unroll
    for (int rt = 0; rt < 4; ++rt) {
      const float* bp = Bv + t * NST + rt * 16 + (low ? 0 : 8);
      v4f b0 = *(const v4f*)bp;
      v4f b1 = *(const v4f*)(bp + 4);
      v8f d;
      d[0] = b0.x * u; d[1] = b0.y * u; d[2] = b0.z * u; d[3] = b0.w * u;
      d[4] = b1.x * u; d[5] = b1.y * u; d[6] = b1.z * u; d[7] = b1.w * u;
      D[ct][rt] = d;
    }
  }
}

// Store D tiles to a [b][n] (b*64+n) laid-out buffer: two b128 per tile/lane.
__device__ __forceinline__ void store_tiles(float* __restrict__ dst,
                                            v8f D[2][4], int colbase, int l15,
                                            bool low) {
#pragma unroll
  for (int ct = 0; ct < 2; ++ct) {
    int b = colbase + ct * 16 + l15;
#pragma unroll
    for (int rt = 0; rt < 4; ++rt) {
      float* p = dst + b * NST + rt * 16 + (low ? 0 : 8);
      v8f d = D[ct][rt];
      v4f lo, hi;
      lo.x = d[0]; lo.y = d[1]; lo.z = d[2]; lo.w = d[3];
      hi.x = d[4]; hi.y = d[5]; hi.z = d[6]; hi.w = d[7];
      *(v4f*)p = lo;
      *(v4f*)(p + 4) = hi;
    }
  }
}

__device__ __forceinline__ void load_tiles(const float* __restrict__ src,
                                           v8f D[2][4], int colbase, int l15,
                                           bool low) {
#pragma unroll
  for (int ct = 0; ct < 2; ++ct) {
    int b = colbase + ct * 16 + l15;
#pragma unroll
    for (int rt = 0; rt < 4; ++rt) {
      const float* p = src + b * NST + rt * 16 + (low ? 0 : 8);
      v4f lo = *(const v4f*)p;
      v4f hi = *(const v4f*)(p + 4);
      v8f d;
      d[0] = lo.x; d[1] = lo.y; d[2] = lo.z; d[3] = lo.w;
      d[4] = hi.x; d[5] = hi.y; d[6] = hi.z; d[7] = hi.w;
      D[ct][rt] = d;
    }
  }
}

// ---------------- Phase P: per-chunk cumulative A product -------------------
__global__ void __launch_bounds__(64) hippo_phaseP(const float* __restrict__ A,
                                                   float* __restrict__ ws_P) {
  __shared__ float Abuf[2][NST * APITCH];
  int c = blockIdx.x;
  int tid = threadIdx.x;
  int wave = tid >> 5, lane = tid & 31;
  int l15 = lane & 15;
  bool low = lane < 16;
  int kof = low ? 0 : 2;
  int colbase = wave * 32;  // columns of P (0..63)
  unsigned lds0 = (unsigned)(uintptr_t)&Abuf[0][0];

  v2f Xb[2][16];
  v8f D[2][4];
#pragma unroll
  for (int ct = 0; ct < 2; ++ct)
#pragma unroll
    for (int kc = 0; kc < 16; ++kc) {  // X = Identity
      int col = colbase + ct * 16 + l15;
      v2f f;
      f.x = (4 * kc + kof == col) ? 1.0f : 0.0f;
      f.y = (4 * kc + 1 + kof == col) ? 1.0f : 0.0f;
      Xb[ct][kc] = f;
    }

  stage_A_async<64>(A + (size_t)(c * TCH) * 4096, lds0, tid);
  for (int tt = 0; tt < TCH; ++tt) {
    wait_async_all();
    __syncthreads();
    if (tt + 1 < TCH)
      stage_A_async<64>(A + (size_t)(c * TCH + tt + 1) * 4096,
                        lds0 + (unsigned)(((tt + 1) & 1) * ABUFB), tid);
#pragma unroll
    for (int ct = 0; ct < 2; ++ct)
#pragma unroll
      for (int rt = 0; rt < 4; ++rt) D[ct][rt] = (v8f)(0.0f);
    scan_step_lds(&Abuf[tt & 1][0], Xb, D, l15, kof);
    d_to_x(D, Xb, low);
  }
  // store P_c row-major [n][j]
#pragma unroll
  for (int ct = 0; ct < 2; ++ct) {
    int col = colbase + ct * 16 + l15;
#pragma unroll
    for (int rt = 0; rt < 4; ++rt)
#pragma unroll
      for (int r = 0; r < 8; ++r)
        ws_P[c * 4096 + (rt * 16 + r + (low ? 0 : 8)) * NST + col] =
            D[ct][rt][r];
  }
}

// ---------------- Phase A: local scan, keep only y_last ---------------------
__global__ void __launch_bounds__(256) hippo_phaseA(
    const float* __restrict__ inp, const float* __restrict__ A,
    const float* __restrict__ Bv, float* __restrict__ ws_y) {
  __shared__ float Abuf[2][NST * APITCH];
  int c = blockIdx.x;
  int tid = threadIdx.x;
  int wave = tid >> 5, lane = tid & 31;
  int l15 = lane & 15;
  bool low = lane < 16;
  int kof = low ? 0 : 2;
  int colbase = wave * 32;
  unsigned lds0 = (unsigned)(uintptr_t)&Abuf[0][0];

  v2f Xb[2][16];
  v8f D[2][4];
#pragma unroll
  for (int ct = 0; ct < 2; ++ct)
#pragma unroll
    for (int kc = 0; kc < 16; ++kc) Xb[ct][kc] = (v2f)(0.0f);

  stage_A_async<256>(A + (size_t)(c * TCH) * 4096, lds0, tid);
  for (int tt = 0; tt < TCH; ++tt) {
    int t = c * TCH + tt;
    wait_async_all();
    __syncthreads();
    if (tt + 1 < TCH)
      stage_A_async<256>(A + (size_t)(t + 1) * 4096,
                         lds0 + (unsigned)(((tt + 1) & 1) * ABUFB), tid);
    init_u(inp, Bv, t, colbase, l15, low, D);
    scan_step_lds(&Abuf[tt & 1][0], Xb, D, l15, kof);
    d_to_x(D, Xb, low);
  }
  store_tiles(ws_y + c * (NB * NST), D, colbase, l15, low);  // y_last(c)
}

// ---------------- Phase B: serial carry over chunk boundaries ---------------
__global__ void __launch_bounds__(256) hippo_phaseB(
    const float* __restrict__ ws_P, const float* __restrict__ ws_y,
    float* __restrict__ ws_s) {
  int wave = threadIdx.x >> 5, lane = threadIdx.x & 31;
  int l15 = lane & 15;
  bool low = lane < 16;
  int kof = low ? 0 : 2;
  int colbase = wave * 32;

  v2f Xb[2][16];  // s_c
  v8f D[2][4];
#pragma unroll
  for (int ct = 0; ct < 2; ++ct)
#pragma unroll
    for (int kc = 0; kc < 16; ++kc) Xb[ct][kc] = (v2f)(0.0f);

  for (int c = 0; c < CHUNKS; ++c) {
    if (c + 1 < CHUNKS) __builtin_prefetch(ws_P + (c + 1) * 4096, 0, 0);
    // publish s_c for phase C, layout [n][b]
    float* sd = ws_s + c * (NB * NST);
#pragma unroll
    for (int ct = 0; ct < 2; ++ct) {
      int col = colbase + ct * 16 + l15;
#pragma unroll
      for (int kc = 0; kc < 16; ++kc) {
        sd[(4 * kc + kof) * NB + col] = Xb[ct][kc].x;
        sd[(4 * kc + 1 + kof) * NB + col] = Xb[ct][kc].y;
      }
    }
    // s_{c+1} = P_c @ s_c + y_last(c)
    load_tiles(ws_y + c * (NB * NST), D, colbase, l15, low);
    scan_step(ws_P + c * 4096, Xb, D, l15, kof);
    d_to_x(D, Xb, low);
  }
}

// ---------------- Phase C: full scan per chunk from s_c ---------------------
__global__ void __launch_bounds__(256) hippo_phaseC(
    const float* __restrict__ inp, const float* __restrict__ A,
    const float* __restrict__ Bv, const float* __restrict__ ws_s,
    float* __restrict__ out) {
  __shared__ float Abuf[2][NST * APITCH];
  int c = blockIdx.x;
  int tid = threadIdx.x;
  int wave = tid >> 5, lane = tid & 31;
  int l15 = lane & 15;
  bool low = lane < 16;
  int kof = low ? 0 : 2;
  int colbase = wave * 32;
  unsigned lds0 = (unsigned)(uintptr_t)&Abuf[0][0];

  v2f Xb[2][16];
  v8f D[2][4];
  const float* s = ws_s + c * (NB * NST);
#pragma unroll
  for (int ct = 0; ct < 2; ++ct) {
    int col = colbase + ct * 16 + l15;
#pragma unroll
    for (int kc = 0; kc < 16; ++kc) {
      v2f f;
      f.x = s[(4 * kc + kof) * NB + col];
      f.y = s[(4 * kc + 1 + kof) * NB + col];
      Xb[ct][kc] = f;
    }
  }

  stage_A_async<256>(A + (size_t)(c * TCH) * 4096, lds0, tid);
  for (int tt = 0; tt < TCH; ++tt) {
    int t = c * TCH + tt;
    wait_async_all();
    __syncthreads();
    if (tt + 1 < TCH)
      stage_A_async<256>(A + (size_t)(t + 1) * 4096,
                         lds0 + (unsigned)(((tt + 1) & 1) * ABUFB), tid);
    init_u(inp, Bv, t, colbase, l15, low, D);
    scan_step_lds(&Abuf[tt & 1][0], Xb, D, l15, kof);
    store_tiles(out + (size_t)t * (NB * NST), D, colbase, l15, low);
    d_to_x(D, Xb, low);
  }
}

extern "C" void kernel_launch(void* const* d_in, const int* in_sizes, int n_in,
                              void* d_out, int out_size, void* d_ws,
                              size_t ws_size, hipStream_t stream) {
  (void)in_sizes; (void)n_in; (void)out_size; (void)ws_size;
  const float* inp = (const float*)d_in[0];  // (L, B)
  const float* A   = (const float*)d_in[1];  // (L, N, N)
  const float* Bv  = (const float*)d_in[2];  // (L, N)
  float* out = (float*)d_out;                // (L, B, N)

  float* wsf  = (float*)d_ws;
  float* ws_P = wsf;                          // 64 * 4096  floats (1 MB)
  float* ws_y = ws_P + CHUNKS * 4096;         // 64 * 16384 floats (4 MB)
  float* ws_s = ws_y + CHUNKS * (NB * NST);   // 64 * 16384 floats (4 MB)

  hippo_phaseP<<<CHUNKS, 64, 0, stream>>>(A, ws_P);
  hippo_phaseA<<<CHUNKS, 256, 0, stream>>>(inp, A, Bv, ws_y);
  hippo_phaseB<<<1, 256, 0, stream>>>(ws_P, ws_y, ws_s);
  hippo_phaseC<<<CHUNKS, 256, 0, stream>>>(inp, A, Bv, ws_s, out);
}